// Sage_64433099375268
// MI455X (gfx1250) — compile-verified
//
#include <hip/hip_runtime.h>

// GraphSAGE forward for MI455X (gfx1250, wave32, WMMA).
// Memory-bound (~350MB compulsory HBM traffic @ 23.3TB/s ~= 15us floor);
// intermediates (h ping-pong + agg, ~77MB) live in the 192MB L2.
// All GEMMs use full-precision V_WMMA_F32_16X16X4_F32; B matrices are
// repacked into fragment order (K-pairs interleaved) so every WMMA's B
// operand is a single b64 load (global or LDS) instead of two b32 loads.

#define N_NODES 100000
#define NFEAT   512
#define HDIM    64
#define NCLASS  256
#define NEDGE   3200000

typedef __attribute__((ext_vector_type(2))) float v2f;
typedef __attribute__((ext_vector_type(4))) float v4f;
typedef __attribute__((ext_vector_type(8))) float v8f;

static __device__ __forceinline__ v8f wmma_f32(v2f a, v2f b, v8f c) {
    // 8 args: (neg_a, A, neg_b, B, c_mod, C, reuse_a, reuse_b)
    return __builtin_amdgcn_wmma_f32_16x16x4_f32(false, a, false, b, (short)0, c,
                                                 false, false);
}

// ---------------------------------------------------------------- degree prep
__global__ void sage_zero_u32(unsigned* __restrict__ p, int n) {
    int i = blockIdx.x * blockDim.x + threadIdx.x;
    if (i < n) p[i] = 0u;
}

__global__ void sage_count_deg(const int* __restrict__ dst, unsigned* __restrict__ cnt) {
    int e = blockIdx.x * blockDim.x + threadIdx.x;
    if (e < NEDGE) atomicAdd(&cnt[dst[e]], 1u);
}

__global__ void sage_make_inv(float* __restrict__ inv, int n) {
    int i = blockIdx.x * blockDim.x + threadIdx.x;
    if (i < n) {
        unsigned c = ((const unsigned*)inv)[i];
        inv[i] = 1.0f / (float)(c > 1u ? c : 1u);   // clamp >= 1 as in PyG
    }
}

__global__ void sage_zero_f4(float* __restrict__ p, int n4) {
    int i = blockIdx.x * blockDim.x + threadIdx.x;
    if (i < n4) ((v4f*)p)[i] = (v4f){0.f, 0.f, 0.f, 0.f};
}

// ------------------------------------------------- weight fragment repacking
// P[p][col] = { W[2p][col], W[2p+1][col] }  (p = K-pair index, row-major cols)
// -> a WMMA B-fragment for K-step k, half h is the single float2 at
//    P[(k/2 + h)*cols + ncol].
__global__ void sage_pack_pairs(const float* __restrict__ W, v2f* __restrict__ P,
                                int kdim, int cols) {
    int i = blockIdx.x * blockDim.x + threadIdx.x;        // over (kdim/2)*cols
    if (i < (kdim / 2) * cols) {
        int p = i / cols, col = i % cols;
        v2f v;
        v.x = W[(2 * p) * cols + col];
        v.y = W[(2 * p + 1) * cols + col];
        P[p * cols + col] = v;
    }
}

// ------------------------------------------------------------- edge scatter
// 64 consecutive threads per edge: coalesced 256B gather of h[src] and
// coalesced fp32 atomics into agg[dst]; both h and agg are L2-resident.
__global__ void sage_scatter(const float* __restrict__ h,
                             const int* __restrict__ src,
                             const int* __restrict__ dst,
                             float* __restrict__ agg) {
    int tid = blockIdx.x * blockDim.x + threadIdx.x;
    int e = tid >> 6;
    int f = tid & 63;
    if (e < NEDGE) {
        float v = h[(long long)src[e] * HDIM + f];
        atomicAdd(&agg[(long long)dst[e] * HDIM + f], v);
    }
}

// ---------------------------------------------------------- h = x@Win + b_in
// 128 threads (4 waves) per 16-row tile; x tile staged in LDS (stride 516 ->
// conflict-free A-frag b64 reads); B frags are single b64 loads from packed
// WinP (128KB, L2-resident). Each wave owns one 16-col strip of HDIM=64.
__global__ void sage_gemm_in(const float* __restrict__ x,
                             const v2f* __restrict__ WinP,
                             const float* __restrict__ bin,
                             float* __restrict__ h) {
    constexpr int KS = NFEAT + 4;           // 516: (4*row + k) % 64 -> no conflicts
    __shared__ float ldsX[16 * KS];
    const int tid  = threadIdx.x;
    const int row0 = blockIdx.x * 16;

    for (int i = tid; i < 16 * (NFEAT / 4); i += 128) {
        int r  = i / (NFEAT / 4);
        int c4 = (i % (NFEAT / 4)) * 4;
        v4f v = *(const v4f*)(x + (long long)(row0 + r) * NFEAT + c4);
        *(v4f*)(ldsX + r * KS + c4) = v;    // (r*516 + c4)*4 is 16B aligned
    }
    __syncthreads();

    const int wave = tid >> 5, lane = tid & 31;
    const int half = lane >> 4, l15 = lane & 15;
    const int ncol = wave * 16 + l15;

    v8f c;
    const float bv = bin[ncol];
#pragma unroll
    for (int r = 0; r < 8; ++r) c[r] = bv;

    for (int k = 0; k < NFEAT; k += 4) {
        v2f a = *(const v2f*)(ldsX + l15 * KS + k + 2 * half);
        v2f b = WinP[(k / 2 + half) * HDIM + ncol];      // one b64 per WMMA
        c = wmma_f32(a, b, c);
    }

#pragma unroll
    for (int r = 0; r < 8; ++r) {
        int m = r + 8 * half;                            // C/D layout: M = r + 8*hi
        h[(long long)(row0 + m) * HDIM + ncol] = c[r];
    }
}

// ---------------- h_out = h + relu( (agg*inv)@Wl + bl + h@Wr ) --------------
// Wl/Wr staged in LDS in fragment-paired float2 layout with row stride 80
// (2*80 % 64 == 32 -> half-wave lane groups hit disjoint 32-bank halves, so
// B-frag ds_load_b64 is conflict-free). agg tile is scaled by inv_cnt while
// staging; both GEMMs accumulate into one WMMA accumulator; residual h is
// re-read from LDS.
__global__ void sage_layer(const float* __restrict__ hin,
                           const float* __restrict__ agg,
                           const float* __restrict__ inv,
                           const float* __restrict__ Wl,
                           const float* __restrict__ bl,
                           const float* __restrict__ Wr,
                           float* __restrict__ hout) {
    constexpr int KS = HDIM + 4;            // 68 for A/H tiles (float stride)
    constexpr int PS = 80;                  // float2 stride for packed weights
    __shared__ v2f   ldsWl[(HDIM / 2) * PS];
    __shared__ v2f   ldsWr[(HDIM / 2) * PS];
    __shared__ float ldsA[16 * KS];
    __shared__ float ldsH[16 * KS];
    const int tid  = threadIdx.x;
    const int row0 = blockIdx.x * 16;

    // stage weights fragment-paired: ldsW[p*PS + col] = {W[2p][col], W[2p+1][col]}
    for (int i = tid; i < (HDIM / 2) * HDIM; i += 128) {
        int p = i / HDIM, col = i % HDIM;
        v2f wl, wr;
        wl.x = Wl[(2 * p) * HDIM + col];  wl.y = Wl[(2 * p + 1) * HDIM + col];
        wr.x = Wr[(2 * p) * HDIM + col];  wr.y = Wr[(2 * p + 1) * HDIM + col];
        ldsWl[p * PS + col] = wl;
        ldsWr[p * PS + col] = wr;
    }
    for (int i = tid; i < 16 * (HDIM / 4); i += 128) {
        int r  = i / (HDIM / 4);
        int c4 = (i % (HDIM / 4)) * 4;
        float s = inv[row0 + r];
        v4f va = *(const v4f*)(agg + (long long)(row0 + r) * HDIM + c4);
        va.x *= s; va.y *= s; va.z *= s; va.w *= s;     // fused scatter-mean scale
        *(v4f*)(ldsA + r * KS + c4) = va;
        *(v4f*)(ldsH + r * KS + c4) =
            *(const v4f*)(hin + (long long)(row0 + r) * HDIM + c4);
    }
    __syncthreads();

    const int wave = tid >> 5, lane = tid & 31;
    const int half = lane >> 4, l15 = lane & 15;
    const int ncol = wave * 16 + l15;

    v8f c;
    const float bv = bl[ncol];
#pragma unroll
    for (int r = 0; r < 8; ++r) c[r] = bv;

#pragma unroll
    for (int k = 0; k < HDIM; k += 4) {
        v2f a  = *(const v2f*)(ldsA + l15 * KS + k + 2 * half);
        v2f b  = ldsWl[(k / 2 + half) * PS + ncol];
        c = wmma_f32(a, b, c);                           // agg @ Wl

        v2f a2 = *(const v2f*)(ldsH + l15 * KS + k + 2 * half);
        v2f b2 = ldsWr[(k / 2 + half) * PS + ncol];
        c = wmma_f32(a2, b2, c);                         // + h @ Wr
    }

#pragma unroll
    for (int r = 0; r < 8; ++r) {
        int m = r + 8 * half;
        float conv = c[r];
        float res  = ldsH[m * KS + ncol] + (conv > 0.f ? conv : 0.f);
        hout[(long long)(row0 + m) * HDIM + ncol] = res;
    }
}

// ------------------------------------------------------ out = h@Wout + b_out
__global__ void sage_gemm_out(const float* __restrict__ h,
                              const v2f* __restrict__ WoutP,
                              const float* __restrict__ bout,
                              float* __restrict__ out) {
    constexpr int KS = HDIM + 4;
    __shared__ float ldsH[16 * KS];
    const int tid  = threadIdx.x;
    const int row0 = blockIdx.x * 16;

    for (int i = tid; i < 16 * (HDIM / 4); i += 128) {
        int r  = i / (HDIM / 4);
        int c4 = (i % (HDIM / 4)) * 4;
        *(v4f*)(ldsH + r * KS + c4) =
            *(const v4f*)(h + (long long)(row0 + r) * HDIM + c4);
    }
    __syncthreads();

    const int wave = tid >> 5, lane = tid & 31;
    const int half = lane >> 4, l15 = lane & 15;

    for (int nt = wave; nt < NCLASS / 16; nt += 4) {     // 4 N-tiles per wave
        const int ncol = nt * 16 + l15;
        v8f c;
        const float bv = bout[ncol];
#pragma unroll
        for (int r = 0; r < 8; ++r) c[r] = bv;

#pragma unroll
        for (int k = 0; k < HDIM; k += 4) {
            v2f a = *(const v2f*)(ldsH + l15 * KS + k + 2 * half);
            v2f b = WoutP[(k / 2 + half) * NCLASS + ncol];   // one b64 per WMMA
            c = wmma_f32(a, b, c);
        }
#pragma unroll
        for (int r = 0; r < 8; ++r) {
            int m = r + 8 * half;
            out[(long long)(row0 + m) * NCLASS + ncol] = c[r];
        }
    }
}

// --------------------------------------------------------------------- host
extern "C" void kernel_launch(void* const* d_in, const int* in_sizes, int n_in,
                              void* d_out, int out_size, void* d_ws, size_t ws_size,
                              hipStream_t stream) {
    const float* x     = (const float*)d_in[0];
    const int*   ei    = (const int*)  d_in[1];   // [2, E]: row0=src, row1=dst
    const float* Win   = (const float*)d_in[2];
    const float* b_in  = (const float*)d_in[3];
    const float* Wl    = (const float*)d_in[4];   // [4,64,64]
    const float* bl    = (const float*)d_in[5];   // [4,64]
    const float* Wr    = (const float*)d_in[6];   // [4,64,64]
    const float* Wout  = (const float*)d_in[7];
    const float* b_out = (const float*)d_in[8];
    float*       out   = (float*)d_out;

    const int* src = ei;
    const int* dst = ei + NEDGE;

    // workspace carve-up (fp32 units):
    //   inv_cnt | h0 | h1 | agg | WinP (packed) | WoutP (packed)   (~77.5 MB)
    float* ws    = (float*)d_ws;
    float* inv   = ws;                                 // N floats (u32 cnt first)
    float* h0    = ws + 131072;
    float* h1    = h0 + (long long)N_NODES * HDIM;
    float* agg   = h1 + (long long)N_NODES * HDIM;
    v2f*   WinP  = (v2f*)(agg + (long long)N_NODES * HDIM);      // 256*64 v2f
    v2f*   WoutP = WinP + (NFEAT / 2) * HDIM;                    // 32*256 v2f

    const int MTILES = N_NODES / 16;                   // 100000/16 = 6250 exactly

    sage_zero_u32 <<<(N_NODES + 255) / 256, 256, 0, stream>>>((unsigned*)inv, N_NODES);
    sage_count_deg<<<(NEDGE   + 255) / 256, 256, 0, stream>>>(dst, (unsigned*)inv);
    sage_make_inv <<<(N_NODES + 255) / 256, 256, 0, stream>>>(inv, N_NODES);

    sage_pack_pairs<<<((NFEAT / 2) * HDIM   + 255) / 256, 256, 0, stream>>>(
        Win, WinP, NFEAT, HDIM);
    sage_pack_pairs<<<((HDIM / 2) * NCLASS + 255) / 256, 256, 0, stream>>>(
        Wout, WoutP, HDIM, NCLASS);

    sage_gemm_in<<<MTILES, 128, 0, stream>>>(x, WinP, b_in, h0);

    float* hin  = h0;
    float* hout = h1;
    for (int i = 0; i < 4; ++i) {
        const int n4 = N_NODES * HDIM / 4;
        sage_zero_f4<<<(n4 + 255) / 256, 256, 0, stream>>>(agg, n4);
        sage_scatter<<<(NEDGE * 64) / 256, 256, 0, stream>>>(hin, src, dst, agg);
        sage_layer<<<MTILES, 128, 0, stream>>>(hin, agg, inv,
                                               Wl + i * HDIM * HDIM, bl + i * HDIM,
                                               Wr + i * HDIM * HDIM, hout);
        float* t = hin; hin = hout; hout = t;
    }
    // after 4 swaps the final h lives in h0 == hin
    sage_gemm_out<<<MTILES, 128, 0, stream>>>(hin, WoutP, b_out, out);
}